// TransitionParser_19207093748352
// MI455X (gfx1250) — compile-verified
//
#include <hip/hip_runtime.h>
#include <hip/hip_bf16.h>

// ---------------------------------------------------------------------------
// MI455X plan:
//  Phase 1: persistent wave32 kernel runs the T=64 sequential recurrence
//           (LSTM gates, q, attention ctx) with software grid barriers
//           (global atomics + s_sleep spin). Recurrent weights (~76MB) are
//           L2-resident (192MB L2), so each step is L2-BW bound.
//  Phase 2: logits heads are batched over all 64 steps into real GEMMs
//           (M=64) executed with V_WMMA_F32_16X16X4_F32, so Ww2 (123MB) is
//           streamed from HBM exactly once (~5.3us at 23.3TB/s) instead of 64x.
//           Each wave owns a 16x64 strip: 4 independent wmma chains sharing
//           one A fragment; edge columns are CLAMPED (in-bounds loads,
//           garbage dropped by guarded stores) -> branch-free inner loop;
//           software-pipelined with unrolling DISABLED so the compiler keeps
//           the pipeline stage intact (partial loadcnt waits, 5 loads in
//           flight per wave) instead of re-serializing via register reuse.
// ---------------------------------------------------------------------------

#define HDIM 1024
#define DDIM 1024
#define SLEN 512
#define TSTEPS 64
#define VA 30
#define VW 30000

#define REC_WGS 256
#define REC_TPB 256
#define REC_NT (REC_WGS * REC_TPB)
#define REC_NWAVES (REC_NT / 32)

// workspace layout (float offsets)
#define OFF_STATE 0                       // [2][5][1024] h_term,c_term,h_act,c_act,stack
#define OFF_GATES (OFF_STATE + 10240)     // [8192] act gates [0,4096), term [4096,8192)
#define OFF_Q     (OFF_GATES + 8192)      // [1024]
#define OFF_FINAL (OFF_Q + 1024)          // [64][4096] = [states(3072), ctx(1024)]
#define OFF_H1A   (OFF_FINAL + 64 * 4096) // [64][1024]
#define OFF_H1W   (OFF_H1A + 64 * 1024)   // [64][1024]
#define OFF_BAR   (OFF_H1W + 64 * 1024)   // 2 x u32 barrier words
// total ~1.66 MB of d_ws

typedef __attribute__((ext_vector_type(2))) float v2f;
typedef __attribute__((ext_vector_type(8))) float v8f;

__device__ __forceinline__ float sigf(float x) { return 1.f / (1.f + __expf(-x)); }

__device__ __forceinline__ float wave_reduce_bcast(float v) {
  for (int o = 16; o; o >>= 1) v += __shfl_down(v, o, 32);
  return __shfl(v, 0, 32);
}

// sense-reversing grid barrier over global memory (all REC_WGS groups resident)
__device__ __forceinline__ void grid_sync(unsigned* bar) {
  __threadfence();
  __syncthreads();
  if (threadIdx.x == 0) {
    unsigned gen = __hip_atomic_load(&bar[1], __ATOMIC_ACQUIRE, __HIP_MEMORY_SCOPE_AGENT);
    unsigned arrived =
        __hip_atomic_fetch_add(&bar[0], 1u, __ATOMIC_ACQ_REL, __HIP_MEMORY_SCOPE_AGENT);
    if (arrived == (unsigned)REC_WGS - 1u) {
      __hip_atomic_store(&bar[0], 0u, __ATOMIC_RELAXED, __HIP_MEMORY_SCOPE_AGENT);
      __hip_atomic_fetch_add(&bar[1], 1u, __ATOMIC_RELEASE, __HIP_MEMORY_SCOPE_AGENT);
    } else {
      while (__hip_atomic_load(&bar[1], __ATOMIC_ACQUIRE, __HIP_MEMORY_SCOPE_AGENT) == gen)
        __builtin_amdgcn_s_sleep(2);
    }
  }
  __syncthreads();
}

// zero states + gates + q + FINAL (ctx is atomically accumulated), reset barrier
__global__ void init_ws_kernel(float* ws) {
  int gid = blockIdx.x * blockDim.x + threadIdx.x;
  const int nz = OFF_H1A;  // zero everything up to H1A (states..FINAL)
  for (int i = gid; i < nz; i += gridDim.x * blockDim.x) ws[i] = 0.f;
  if (gid == 0) {
    unsigned* bar = (unsigned*)(ws + OFF_BAR);
    bar[0] = 0u;
    bar[1] = 0u;
  }
}

__global__ void __launch_bounds__(REC_TPB) recurrence_kernel(
    const float* __restrict__ memory, const int* __restrict__ actions,
    const int* __restrict__ words,
    const float* __restrict__ tWih, const float* __restrict__ tWhh,
    const float* __restrict__ tbih, const float* __restrict__ tbhh,
    const float* __restrict__ aWih, const float* __restrict__ aWhh,
    const float* __restrict__ abih, const float* __restrict__ abhh,
    const float* __restrict__ Wq, const float* __restrict__ bq,
    const float* __restrict__ act_emb, const float* __restrict__ word_emb,
    float* __restrict__ ws) {
  const int gid = blockIdx.x * REC_TPB + threadIdx.x;
  const int lane = threadIdx.x & 31;
  const int wid = gid >> 5;

  float* states = ws + OFF_STATE;
  float* gates = ws + OFF_GATES;
  float* qv = ws + OFF_Q;
  float* FINALB = ws + OFF_FINAL;
  unsigned* bar = (unsigned*)(ws + OFF_BAR);

  for (int t = 0; t < TSTEPS; ++t) {
    const int cur = t & 1, nxt = cur ^ 1;
    float* st_c = states + cur * 5 * HDIM;
    float* st_n = states + nxt * 5 * HDIM;
    const int a_t = actions[t];
    const int w_t = words[t];
    const float* a_e = act_emb + (long)a_t * HDIM;
    const float* w_e = word_emb + (long)w_t * HDIM;
    const float* h_term = st_c + 0 * HDIM;
    const float* c_term = st_c + 1 * HDIM;
    const float* h_act = st_c + 2 * HDIM;
    const float* c_act = st_c + 3 * HDIM;
    const float* stk = st_c + 4 * HDIM;

    // ---- Stage A: one wave per dot product (gate rows + q rows) ----
    for (int task = wid; task < 9 * HDIM; task += REC_NWAVES) {
      float sum = 0.f;
      if (task < 4 * HDIM) {  // act LSTM gate row
        const int r = task;
        const float* wi = aWih + (long)r * HDIM;
        const float* wh = aWhh + (long)r * HDIM;
        for (int k = lane; k < HDIM; k += 32) sum += wi[k] * a_e[k] + wh[k] * h_act[k];
        sum = wave_reduce_bcast(sum);
        if (lane == 0) gates[r] = sum + abih[r] + abhh[r];
      } else if (task < 8 * HDIM) {  // term LSTM gate row
        const int r = task - 4 * HDIM;
        const float* wi = tWih + (long)r * HDIM;
        const float* wh = tWhh + (long)r * HDIM;
        for (int k = lane; k < HDIM; k += 32) sum += wi[k] * w_e[k] + wh[k] * h_term[k];
        sum = wave_reduce_bcast(sum);
        if (lane == 0) gates[4 * HDIM + r] = sum + tbih[r] + tbhh[r];
      } else {  // q row: states = [h_term, h_act, stack]
        const int d = task - 8 * HDIM;
        const float* wq = Wq + (long)d * (3 * HDIM);
        for (int k = lane; k < 3 * HDIM; k += 32) {
          float x = (k < HDIM) ? h_term[k]
                               : (k < 2 * HDIM) ? h_act[k - HDIM] : stk[k - 2 * HDIM];
          sum += wq[k] * x;
        }
        sum = wave_reduce_bcast(sum);
        if (lane == 0) qv[d] = sum + bq[d];
      }
    }
    grid_sync(bar);

    // ---- Stage B: LSTM updates, stack<-w_e, FINAL states, attention ctx ----
    for (int u = gid; u < 6 * HDIM; u += REC_NT) {
      if (u < 2 * HDIM) {  // 2048 LSTM units (act then term)
        const int j = (u < HDIM) ? u : u - HDIM;
        const float* g = (u < HDIM) ? gates : gates + 4 * HDIM;
        const float* cp = (u < HDIM) ? c_act : c_term;
        const float gi = g[j], gf = g[j + HDIM], gg = g[j + 2 * HDIM], go = g[j + 3 * HDIM];
        const float c2 = sigf(gf) * cp[j] + sigf(gi) * tanhf(gg);
        const float h2 = sigf(go) * tanhf(c2);
        if (u < HDIM) { st_n[2 * HDIM + j] = h2; st_n[3 * HDIM + j] = c2; }
        else          { st_n[0 * HDIM + j] = h2; st_n[1 * HDIM + j] = c2; }
      } else if (u < 3 * HDIM) {  // stack carry <- w_e
        const int j = u - 2 * HDIM;
        st_n[4 * HDIM + j] = w_e[j];
      } else {  // FINAL[t][0..3071] = entry states
        const int j = u - 3 * HDIM;
        const float x = (j < HDIM) ? h_term[j]
                                   : (j < 2 * HDIM) ? h_act[j - HDIM] : stk[j - 2 * HDIM];
        FINALB[(long)t * 4096 + j] = x;
      }
    }
    // attention: one wave per memory row; ctx accumulated via f32 atomics
    float* ctx = FINALB + (long)t * 4096 + 3072;
    for (int s = wid; s < SLEN; s += REC_NWAVES) {
      const float* mrow = memory + (long)s * DDIM;
      float dot = 0.f;
      for (int k = lane; k < DDIM; k += 32) dot += mrow[k] * qv[k];
      dot = wave_reduce_bcast(dot);
      for (int d = lane; d < DDIM; d += 32) atomicAdd(&ctx[d], dot * mrow[d]);
    }
    grid_sync(bar);
  }
}

// ---------------------------------------------------------------------------
// Phase 2: out[m][n] = act( sum_k A[m][k]*W[n][k] + bias[n] )
// fp32 WMMA 16x16x4; each wave owns a 16(M) x 64(N) strip: 4 accumulator
// chains share one A fragment. Fragment layout per ISA: lane<16 holds
// K={0,1}, lane>=16 holds K={2,3}; A rows m0+lane%16, B cols n0+lane%16
// (both operands K-contiguous row-major -> float2 loads).
// Out-of-range columns clamped to N-1 (results dropped by guarded store).
// Software-pipelined with unroll(disable): each iteration issues 5 loads
// for step k and consumes step k-4's fragments, so wmma only waits for
// older loads (partial loadcnt waits, loads stay in flight).
// ---------------------------------------------------------------------------
#define WMMA_F32(a, b, c) \
  __builtin_amdgcn_wmma_f32_16x16x4_f32(false, (a), false, (b), (short)0, (c), false, false)

__global__ void __launch_bounds__(256) wmma_gemm_nt(
    const float* __restrict__ A,     // [M x K] row-major (M = 64)
    const float* __restrict__ W,     // [N x K] row-major
    const float* __restrict__ bias,  // [N]
    float* __restrict__ out,         // [M x N] row-major
    int M, int N, int K, int relu) {
  const int wave = (blockIdx.x * blockDim.x + threadIdx.x) >> 5;
  const int lane = threadIdx.x & 31;
  const int half = lane >> 4;
  const int l16 = lane & 15;

  const int tiles_m = M >> 4;
  const int quads_n = (N + 63) >> 6;
  const int tm = wave % tiles_m;
  const int tq = wave / tiles_m;
  if (tq >= quads_n) return;

  const int m0 = tm << 4;
  const int nb = tq << 6;  // quad base column
  const int kof = half * 2;

  const int n0 = nb + l16, n1 = n0 + 16, n2 = n0 + 32, n3 = n0 + 48;
  const int c0 = min(n0, N - 1), c1 = min(n1, N - 1);
  const int c2 = min(n2, N - 1), c3 = min(n3, N - 1);
  const float* arow = A + (long)(m0 + l16) * K + kof;
  const float* wr0 = W + (long)c0 * K + kof;
  const float* wr1 = W + (long)c1 * K + kof;
  const float* wr2 = W + (long)c2 * K + kof;
  const float* wr3 = W + (long)c3 * K + kof;

  v8f acc0 = {}, acc1 = {}, acc2 = {}, acc3 = {};

  // pipeline prologue: fragments for k = 0
  v2f ac = *(const v2f*)(arow);
  v2f b0c = *(const v2f*)(wr0);
  v2f b1c = *(const v2f*)(wr1);
  v2f b2c = *(const v2f*)(wr2);
  v2f b3c = *(const v2f*)(wr3);

  // one pipeline stage per iteration; unrolling would let the register
  // allocator collapse the rotation and re-serialize load->wmma.
#pragma clang loop unroll(disable)
  for (int k = 4; k < K; k += 4) {
    v2f an = *(const v2f*)(arow + k);
    v2f b0n = *(const v2f*)(wr0 + k);
    v2f b1n = *(const v2f*)(wr1 + k);
    v2f b2n = *(const v2f*)(wr2 + k);
    v2f b3n = *(const v2f*)(wr3 + k);
    acc0 = WMMA_F32(ac, b0c, acc0);
    acc1 = WMMA_F32(ac, b1c, acc1);
    acc2 = WMMA_F32(ac, b2c, acc2);
    acc3 = WMMA_F32(ac, b3c, acc3);
    ac = an;
    b0c = b0n;
    b1c = b1n;
    b2c = b2n;
    b3c = b3n;
  }
  // pipeline epilogue: last K-step
  acc0 = WMMA_F32(ac, b0c, acc0);
  acc1 = WMMA_F32(ac, b1c, acc1);
  acc2 = WMMA_F32(ac, b2c, acc2);
  acc3 = WMMA_F32(ac, b3c, acc3);

  // D layout: VGPR v, lane -> (m = m0 + v + 8*half, n = tile base + lane%16)
#pragma unroll
  for (int v = 0; v < 8; ++v) {
    const int m = m0 + v + 8 * half;
    float* orow = out + (long)m * N;
    if (n0 < N) {
      float r = acc0[v] + bias[n0];
      if (relu) r = fmaxf(r, 0.f);
      orow[n0] = r;
    }
    if (n1 < N) {
      float r = acc1[v] + bias[n1];
      if (relu) r = fmaxf(r, 0.f);
      orow[n1] = r;
    }
    if (n2 < N) {
      float r = acc2[v] + bias[n2];
      if (relu) r = fmaxf(r, 0.f);
      orow[n2] = r;
    }
    if (n3 < N) {
      float r = acc3[v] + bias[n3];
      if (relu) r = fmaxf(r, 0.f);
      orow[n3] = r;
    }
  }
}

extern "C" void kernel_launch(void* const* d_in, const int* in_sizes, int n_in,
                              void* d_out, int out_size, void* d_ws, size_t ws_size,
                              hipStream_t stream) {
  (void)in_sizes; (void)n_in; (void)out_size; (void)ws_size;
  const float* memory   = (const float*)d_in[0];
  const int*   actions  = (const int*)d_in[1];
  const int*   words    = (const int*)d_in[2];
  const float* tWih = (const float*)d_in[3];
  const float* tWhh = (const float*)d_in[4];
  const float* tbih = (const float*)d_in[5];
  const float* tbhh = (const float*)d_in[6];
  const float* aWih = (const float*)d_in[7];
  const float* aWhh = (const float*)d_in[8];
  const float* abih = (const float*)d_in[9];
  const float* abhh = (const float*)d_in[10];
  const float* Wq  = (const float*)d_in[11];
  const float* bq  = (const float*)d_in[12];
  const float* Wa1 = (const float*)d_in[13];
  const float* ba1 = (const float*)d_in[14];
  const float* Wa2 = (const float*)d_in[15];
  const float* ba2 = (const float*)d_in[16];
  const float* Ww1 = (const float*)d_in[17];
  const float* bw1 = (const float*)d_in[18];
  const float* Ww2 = (const float*)d_in[19];
  const float* bw2 = (const float*)d_in[20];
  const float* act_emb  = (const float*)d_in[21];
  const float* word_emb = (const float*)d_in[22];

  float* ws = (float*)d_ws;
  float* out = (float*)d_out;

  // deterministic re-init of recurrent state / accumulators / barrier
  init_ws_kernel<<<(OFF_H1A + 255) / 256, 256, 0, stream>>>(ws);

  // Phase 1: sequential recurrence (persistent, grid-synced)
  recurrence_kernel<<<REC_WGS, REC_TPB, 0, stream>>>(
      memory, actions, words, tWih, tWhh, tbih, tbhh, aWih, aWhh, abih, abhh,
      Wq, bq, act_emb, word_emb, ws);

  // Phase 2: batched WMMA GEMMs over all 64 steps
  auto launch_gemm = [&](const float* A, const float* W, const float* b, float* o,
                         int M, int N, int K, int relu) {
    const int tiles_m = M >> 4;
    const int quads_n = (N + 63) >> 6;
    const int tasks = tiles_m * quads_n;  // one wave each (16x64 strip)
    const int blocks = (tasks + 7) / 8;   // 8 waves / block
    wmma_gemm_nt<<<blocks, 256, 0, stream>>>(A, W, b, o, M, N, K, relu);
  };

  launch_gemm(ws + OFF_FINAL, Wa1, ba1, ws + OFF_H1A, TSTEPS, HDIM, 4096, 1);
  launch_gemm(ws + OFF_H1A,   Wa2, ba2, out,          TSTEPS, VA,   1024, 0);
  launch_gemm(ws + OFF_FINAL, Ww1, bw1, ws + OFF_H1W, TSTEPS, HDIM, 4096, 1);
  launch_gemm(ws + OFF_H1W,   Ww2, bw2, out + (long)TSTEPS * VA, TSTEPS, VW, 1024, 0);
}